// TriPlaneField_17884243821138
// MI455X (gfx1250) — compile-verified
//
#include <hip/hip_runtime.h>

// ---------------------------------------------------------------------------
// TriPlane field sampling, MI455X (gfx1250, wave32).
// Bandwidth-bound (<1 FLOP/B): WMMA is inapplicable (per-point B operand),
// so the CDNA5 path used here is the Tensor Data Mover in *gather mode*:
//   1) tiled transpose of each grid [3,C,r,r] -> [3,r,r,C] in workspace
//      (b128 loads + b128 stores through a conflict-free 32x33 LDS tile)
//   2) gather kernel: one point per wave32; per (plane,scale) a single
//      TENSOR_LOAD_TO_LDS gather-mode descriptor DMAs the 4 corner rows
//      (4 x 72 contiguous floats) into LDS; a flattened 12-deep software
//      pipeline (issue it+1, s_wait_tensorcnt 1, blend it) keeps the TDM
//      busy end-to-end with two descriptors in flight per wave.
// Falls back to a direct-gather kernel if the workspace is too small.
// ---------------------------------------------------------------------------

#define C_DIM        72
#define NSCALE       4
#define OUTW         (3 * C_DIM * NSCALE)   // 864
#define PTS_PER_BLK  8                      // 8 wave32 per 256-thread block

constexpr int    RES[NSCALE]  = {64, 128, 256, 512};
// float offsets of each transposed scale inside workspace: sum 3*r*r*72
constexpr size_t TOFF[NSCALE] = {0, 884736, 4423680, 18579456};
constexpr size_t TWS_FLOATS   = 75202560;   // total floats (=> 300,810,240 B)

typedef unsigned int u32x4 __attribute__((ext_vector_type(4)));
typedef int          i32x8 __attribute__((ext_vector_type(8)));
typedef int          i32x4 __attribute__((ext_vector_type(4)));

// ---------------------------------------------------------------------------
// Transpose one scale: [3, 72, T] -> [3, T, 72]   (T = r*r)
// grid = (T/32, 3 channel-tiles, 3 planes), block = 256
// ---------------------------------------------------------------------------
__global__ __launch_bounds__(256)
void tp_transpose(const float* __restrict__ src, float* __restrict__ dst, int T)
{
    __shared__ float tile[32][33];           // +1 pad: conflict-free transpose

    const int t0 = blockIdx.x * 32;          // texel tile origin
    const int c0 = blockIdx.y * 32;          // channel tile origin
    const int p  = blockIdx.z;               // plane

    const float* sp = src + (size_t)p * C_DIM * T;
    float*       dp = dst + (size_t)p * (size_t)T * C_DIM;

    const int tid = threadIdx.x;

    // load: 32 channels x 32 texels, one float4 along texels per thread
    {
        const int row = tid >> 3;            // channel within tile (0..31)
        const int col = (tid & 7) * 4;       // texel within tile (0,4,..,28)
        if (c0 + row < C_DIM) {
            const float4 v = *(const float4*)(sp + (size_t)(c0 + row) * T + t0 + col);
            tile[row][col + 0] = v.x;
            tile[row][col + 1] = v.y;
            tile[row][col + 2] = v.z;
            tile[row][col + 3] = v.w;
        }
    }
    __syncthreads();

    // store: 32 texels x 32 channels, one float4 along channels per thread
    {
        const int wr = tid >> 3;             // texel within tile (0..31)
        const int wc = (tid & 7) * 4;        // channel within tile
        if (c0 + wc + 3 < C_DIM) {           // C_DIM % 4 == 0 -> no partials
            float4 o;
            o.x = tile[wc + 0][wr];
            o.y = tile[wc + 1][wr];
            o.z = tile[wc + 2][wr];
            o.w = tile[wc + 3][wr];
            *(float4*)(dp + (size_t)(t0 + wr) * C_DIM + c0 + wc) = o;
        }
    }
}

// ---------------------------------------------------------------------------
// TDM gather-mode issue: fetch 4 corner rows (72 floats each, contiguous in
// the transposed layout) into LDS with one TENSOR_LOAD_TO_LDS.
// D# layout per CDNA5 ISA sect. 8.3-8.7 (gather mode,
// 32-bit row indices, data_size=4B, tile_dim0=tensor_dim0_stride=72).
// ---------------------------------------------------------------------------
__device__ __forceinline__ unsigned rfl(unsigned v)
{
    return (unsigned)__builtin_amdgcn_readfirstlane((int)v);
}

__device__ __forceinline__ void tdm_issue(const float* plane_base, int r,
                                          float cxp, float cyp,
                                          unsigned lds_byte_off)
{
    const float rm1 = (float)(r - 1);
    const float fx  = fminf(fmaxf((cxp + 1.0f) * 0.5f * rm1, 0.0f), rm1);
    const float fy  = fminf(fmaxf((cyp + 1.0f) * 0.5f * rm1, 0.0f), rm1);
    const int x0 = (int)fx, y0 = (int)fy;
    const int x1 = min(x0 + 1, r - 1);
    const int y1 = min(y0 + 1, r - 1);

    const unsigned T   = (unsigned)(r * r);
    const unsigned i00 = rfl((unsigned)(y0 * r + x0));
    const unsigned i01 = rfl((unsigned)(y0 * r + x1));
    const unsigned i10 = rfl((unsigned)(y1 * r + x0));
    const unsigned i11 = rfl((unsigned)(y1 * r + x1));

    const unsigned long long ga = (unsigned long long)(size_t)plane_base;
    const unsigned ga_lo = rfl((unsigned)ga);
    const unsigned ga_hi = rfl((unsigned)(ga >> 32) & 0x01FFFFFFu);

    u32x4 g0;
    g0.x = 0xC0000001u;                 // count=1 | gather_mode | 32-bit idx
    g0.y = rfl(lds_byte_off);           // lds_addr (bytes)
    g0.z = ga_lo;                       // global_addr[31:0]
    g0.w = ga_hi | 0x80000000u;         // global_addr[56:32] | type=2

    i32x8 g1;
    g1[0] = 0x00020000;                 // workgroup_mask=0, data_size=4B
    g1[1] = (int)(72u << 16);           // tensor_dim0 = 72 (row length)
    g1[2] = (int)((T & 0xFFFFu) << 16); // tensor_dim1 lo16 (num rows)
    g1[3] = (int)((T >> 16) | (72u << 16)); // tensor_dim1 hi16 | tile_dim0=72
    g1[4] = 4;                          // tile_dim1 = 4 valid row indices
    g1[5] = 72;                         // tensor_dim0_stride lo32 = 72
    g1[6] = 0;                          // stride hi16 | dim1_stride (ignored)
    g1[7] = 0;

    i32x4 g2;                           // row indices 0..3 (32-bit mode)
    g2[0] = (int)i00; g2[1] = (int)i01; g2[2] = (int)i10; g2[3] = (int)i11;
    i32x4 g3;                           // row indices 4..7 unused
    g3[0] = 0; g3[1] = 0; g3[2] = 0; g3[3] = 0;

    i32x8 g4;                           // extra group (clang-23 6-arg form):
    g4[0] = 0; g4[1] = 0; g4[2] = 0;    // zero-filled
    g4[3] = 0; g4[4] = 0; g4[5] = 0;
    g4[6] = 0; g4[7] = 0;

    __builtin_amdgcn_tensor_load_to_lds(g0, g1, g2, g3, g4, 0);
}

// ---------------------------------------------------------------------------
// Fast path: TDM gather from transposed layout [3, r, r, C].
// One point per wave32; lanes cover channels (c = lane, lane+32, lane+64).
// Flattened 12-iteration pipeline (plane-major, scale-minor), double-buffered:
// issue it+1, wait tensorcnt<=1, blend it from LDS (ds_load path).
// ---------------------------------------------------------------------------
__global__ __launch_bounds__(256)
void tp_sample_tdm(const float* __restrict__ pts,
                   const float* __restrict__ tws,
                   float* __restrict__ out, int N)
{
    __shared__ float tbuf[PTS_PER_BLK][2][4 * C_DIM];   // 18 KB / block

    const int lane = threadIdx.x & 31;
    const int wid  = threadIdx.x >> 5;
    const int pid  = blockIdx.x * PTS_PER_BLK + wid;    // wave-uniform
    if (pid >= N) return;                               // whole wave exits

    // normalize_aabb with aabb [[b,b,b],[-b,-b,-b]] reduces to  pn = -pt / b
    const float k  = -0.625f;                           // -1/1.6
    const float px = pts[3 * pid + 0] * k;
    const float py = pts[3 * pid + 1] * k;
    const float pz = pts[3 * pid + 2] * k;
    const float cx[3] = {px, px, py};
    const float cy[3] = {py, pz, pz};

    const unsigned lds0 = (unsigned)(size_t)&tbuf[wid][0][0]; // escapes into D#
    const unsigned lds1 = (unsigned)(size_t)&tbuf[wid][1][0];

    float* const base = &tbuf[wid][0][0];               // addrspace(3) base

    const size_t outbase = (size_t)pid * OUTW;

    {   // pipeline prologue: fetch (plane 0, scale 0) corners into buffer 0
        const int r0 = RES[0];
        tdm_issue(tws + TOFF[0], r0, cx[0], cy[0], lds0);
    }

    float a0 = 0.0f, a1 = 0.0f, a2 = 0.0f;              // cumsum per chunk

#pragma unroll
    for (int it = 0; it < 3 * NSCALE; ++it) {
        const int p = it >> 2;                          // plane-major
        const int s = it & 3;                           // scale-minor

        if (it + 1 < 3 * NSCALE) {
            const int pn = (it + 1) >> 2;
            const int sn = (it + 1) & 3;
            const int rn = RES[sn];
            tdm_issue(tws + TOFF[sn] + (size_t)pn * rn * rn * C_DIM, rn,
                      cx[pn], cy[pn], ((it + 1) & 1) ? lds1 : lds0);
            __builtin_amdgcn_s_wait_tensorcnt(1);       // iteration it landed
        } else {
            __builtin_amdgcn_s_wait_tensorcnt(0);
        }

        // Opaque 32-bit element offset: keeps accesses in addrspace(3) so
        // they lower to ds_load_b32, while the memory clobber orders them
        // after the tensorcnt wait (TDM wrote LDS behind the compiler).
        unsigned bofs = (unsigned)((it & 1) * 4 * C_DIM);
        asm volatile("" : "+v"(bofs) : : "memory");

        const int   r   = RES[s];
        const float rm1 = (float)(r - 1);
        const float fx  = fminf(fmaxf((cx[p] + 1.0f) * 0.5f * rm1, 0.0f), rm1);
        const float fy  = fminf(fmaxf((cy[p] + 1.0f) * 0.5f * rm1, 0.0f), rm1);
        const float wx  = fx - floorf(fx);
        const float wy  = fy - floorf(fy);
        const float w00 = (1.0f - wx) * (1.0f - wy);
        const float w01 = wx * (1.0f - wy);
        const float w10 = (1.0f - wx) * wy;
        const float w11 = wx * wy;

#pragma unroll
        for (int ci = 0; ci < 3; ++ci) {
            const int c = lane + 32 * ci;
            if (c < C_DIM) {
                const float v = base[bofs + c]             * w00 +
                                base[bofs + C_DIM + c]     * w01 +
                                base[bofs + 2 * C_DIM + c] * w10 +
                                base[bofs + 3 * C_DIM + c] * w11;
                float& a = (ci == 0) ? a0 : (ci == 1) ? a1 : a2;
                const float nv = (s == 0) ? v : (a + v);
                a = nv;
                out[outbase + (size_t)s * (3 * C_DIM) + p * C_DIM + c] = nv;
            }
        }
    }
}

// ---------------------------------------------------------------------------
// Fallback path (workspace too small): gather straight from [3, C, r, r].
// ---------------------------------------------------------------------------
__global__ __launch_bounds__(256)
void tp_sample_direct(const float* __restrict__ pts,
                      const float* __restrict__ g0, const float* __restrict__ g1,
                      const float* __restrict__ g2, const float* __restrict__ g3,
                      float* __restrict__ out, int N)
{
    const int lane = threadIdx.x & 31;
    const int wid  = threadIdx.x >> 5;
    const int pid  = blockIdx.x * PTS_PER_BLK + wid;
    if (pid >= N) return;

    const float k  = -0.625f;
    const float px = pts[3 * pid + 0] * k;
    const float py = pts[3 * pid + 1] * k;
    const float pz = pts[3 * pid + 2] * k;
    const float cx[3] = {px, px, py};
    const float cy[3] = {py, pz, pz};

    const float* G[NSCALE] = {g0, g1, g2, g3};

    float acc[3][3];
    const size_t outbase = (size_t)pid * OUTW;

#pragma unroll
    for (int s = 0; s < NSCALE; ++s) {
        const int    r   = RES[s];
        const float  rm1 = (float)(r - 1);
        const float* g   = G[s];
        const size_t T   = (size_t)r * r;
#pragma unroll
        for (int p = 0; p < 3; ++p) {
            float fx = fminf(fmaxf((cx[p] + 1.0f) * 0.5f * rm1, 0.0f), rm1);
            float fy = fminf(fmaxf((cy[p] + 1.0f) * 0.5f * rm1, 0.0f), rm1);
            const int   x0 = (int)fx;
            const int   y0 = (int)fy;
            const float wx = fx - (float)x0;
            const float wy = fy - (float)y0;
            const int   x1 = min(x0 + 1, r - 1);
            const int   y1 = min(y0 + 1, r - 1);

            const float w00 = (1.0f - wx) * (1.0f - wy);
            const float w01 = wx * (1.0f - wy);
            const float w10 = (1.0f - wx) * wy;
            const float w11 = wx * wy;

#pragma unroll
            for (int ci = 0; ci < 3; ++ci) {
                const int c = lane + 32 * ci;
                if (c < C_DIM) {
                    const float* ch = g + ((size_t)p * C_DIM + c) * T;
                    const float v = ch[(size_t)y0 * r + x0] * w00 +
                                    ch[(size_t)y0 * r + x1] * w01 +
                                    ch[(size_t)y1 * r + x0] * w10 +
                                    ch[(size_t)y1 * r + x1] * w11;
                    const float a = (s == 0) ? v : (acc[p][ci] + v);
                    acc[p][ci] = a;
                    out[outbase + (size_t)s * (3 * C_DIM) + p * C_DIM + c] = a;
                }
            }
        }
    }
}

// ---------------------------------------------------------------------------
extern "C" void kernel_launch(void* const* d_in, const int* in_sizes, int n_in,
                              void* d_out, int out_size, void* d_ws, size_t ws_size,
                              hipStream_t stream)
{
    (void)n_in; (void)out_size;
    const float* pts = (const float*)d_in[0];
    const float* g[NSCALE] = {(const float*)d_in[1], (const float*)d_in[2],
                              (const float*)d_in[3], (const float*)d_in[4]};
    float* out = (float*)d_out;
    const int N = in_sizes[0] / 3;

    const dim3 blk(256);
    const dim3 sgrid((N + PTS_PER_BLK - 1) / PTS_PER_BLK);

    if (ws_size >= TWS_FLOATS * sizeof(float)) {
        float* tws = (float*)d_ws;
        for (int s = 0; s < NSCALE; ++s) {
            const int T = RES[s] * RES[s];
            dim3 tgrid(T / 32, (C_DIM + 31) / 32, 3);
            tp_transpose<<<tgrid, blk, 0, stream>>>(g[s], tws + TOFF[s], T);
        }
        tp_sample_tdm<<<sgrid, blk, 0, stream>>>(pts, tws, out, N);
    } else {
        tp_sample_direct<<<sgrid, blk, 0, stream>>>(pts, g[0], g[1], g[2], g[3],
                                                    out, N);
    }
}